// TuckerMobiusE_89515708383811
// MI455X (gfx1250) — compile-verified
//
#include <hip/hip_runtime.h>
#include <hip/hip_bf16.h>

// ---------------------------------------------------------------------------
// TuckerMobiusE on MI455X (gfx1250, wave32, WMMA)
//
// score[b,n] = sigmoid( sum_e v_a[b,e]*E_a[n,e] + v_b[b,e]*E_b[n,e] + c[b] )
// with bn3 folded into v (scale) / c (bias); bn3 stats recovered from column
// sums S[e] and Gram matrices G = E^T E. No (16,100,50000) intermediate.
//
// WMMA (V_WMMA_F32_16X16X4_F32, full fp32):
//   - Gram G = E^T E   : LDS-staged, E read from HBM exactly once per matrix,
//                        all 49 tiles per block, no guarded loads in hot loop.
//   - score = v @ E^T  : M=16=BATCH, N=50000=625*5*16 exact, K=100=25*4 exact,
//                        v2f (b64) operand loads, no divergence around WMMA.
// ---------------------------------------------------------------------------

typedef __attribute__((ext_vector_type(2))) float v2f;
typedef __attribute__((ext_vector_type(8))) float v8f;

#define NENT   50000
#define DIM    100
#define BATCH  16
#define GPAD   112           // 7*16 padded Gram stride
#define BN_CNT 800000.0f     // BATCH * NENT
#define EPS    1e-5f
#define GSLICE 2000          // K rows per gram block (25 slices exact)
#define GCHUNK 16            // K rows staged in LDS per step

// ---------------- workspace layout (floats) ----------------
#define WS_WMRE   0          // [16*100*100]
#define WS_WMIM   160000     // [16*100*100]
#define WS_SWA    320000     // [100*100]  sum_b Wm_re[b,e,f]
#define WS_SWB    330000     // [100*100]
#define WS_GA     340000     // [112*112]  Gram of E_a (atomic target)
#define WS_GB     352544     // [112*112]
#define WS_SA     365088     // [112]      col sums (atomic target)
#define WS_SB     365200     // [112]
#define WS_QA     365312     // [16*100]   diag(M^T G M) per (b,f)
#define WS_QB     366912     // [16*100]
#define WS_T0     368512     // [16*100]   Mobius real
#define WS_T1     370112     // [16*100]   Mobius imag
#define WS_VA     371712     // [16*100]   folded lhs vectors
#define WS_VB     373312     // [16*100]
#define WS_CV     374912     // [16]       folded bias
#define WS_ZERO_BEG WS_GA
#define WS_ZERO_CNT (2*GPAD*GPAD + 2*GPAD)   // Ga,Gb,Sa,Sb contiguous

// ---------------------------------------------------------------------------
__global__ void k_zero(float* p, int n) {
  int i = blockIdx.x * 256 + threadIdx.x;
  if (i < n) p[i] = 0.0f;
}

// ---------------------------------------------------------------------------
// Wm[b,e,f] = sum_k r2[b,k] * W[k,e,f]; also SW[e,f] = sum_b Wm[b,e,f].
__global__ void k_build_wm(const int* __restrict__ r_idx,
                           const float* __restrict__ R2,
                           const float* __restrict__ Wre,
                           const float* __restrict__ Wim,
                           float* __restrict__ WmRe, float* __restrict__ WmIm,
                           float* __restrict__ SWa,  float* __restrict__ SWb) {
  __shared__ float r2s[BATCH * DIM];
  for (int i = threadIdx.x; i < BATCH * DIM; i += 256) {
    int b = i / DIM, k = i % DIM;
    r2s[i] = R2[r_idx[b] * DIM + k];
  }
  __syncthreads();
  int id = blockIdx.x * 256 + threadIdx.x;     // id = e*100 + f
  if (id >= DIM * DIM) return;
  float ar[BATCH], ai[BATCH];
#pragma unroll
  for (int b = 0; b < BATCH; ++b) { ar[b] = 0.f; ai[b] = 0.f; }
  for (int k = 0; k < DIM; ++k) {
    float wre = Wre[k * DIM * DIM + id];
    float wim = Wim[k * DIM * DIM + id];
#pragma unroll
    for (int b = 0; b < BATCH; ++b) {
      ar[b] = fmaf(r2s[b * DIM + k], wre, ar[b]);
      ai[b] = fmaf(r2s[b * DIM + k], wim, ai[b]);
    }
  }
  float sre = 0.f, sim = 0.f;
#pragma unroll
  for (int b = 0; b < BATCH; ++b) {
    WmRe[b * DIM * DIM + id] = ar[b];
    WmIm[b * DIM * DIM + id] = ai[b];
    sre += ar[b]; sim += ai[b];
  }
  SWa[id] = sre; SWb[id] = sim;
}

// ---------------------------------------------------------------------------
// Gram via WMMA, LDS-staged. grid = (25 K-slices, 1, 2 matrices), block = 256
// (8 waves). Each block stages GCHUNK rows x 112 cols (zero padded) of E into
// LDS, then every wave runs 7 tile-slots x 4 WMMA steps per chunk. 49 tiles
// are covered exactly once (slot ids >= 49 wrap to duplicate tiles whose
// stores are suppressed) so WMMA always executes with EXEC all-ones.
__global__ void k_gram(const float* __restrict__ Ea, const float* __restrict__ Eb,
                       float* __restrict__ Ga, float* __restrict__ Gb) {
  __shared__ float se[GCHUNK * GPAD];
  const float* E = blockIdx.z ? Eb : Ea;
  float*       G = blockIdx.z ? Gb : Ga;
  int tid = threadIdx.x;
  int wave = tid >> 5, lane = tid & 31;
  int half = lane >> 4, lm = lane & 15;
  int n0 = blockIdx.x * GSLICE;

  v8f zero = {};
  v8f acc[7];
#pragma unroll
  for (int t = 0; t < 7; ++t) acc[t] = zero;

  for (int kc = 0; kc < GSLICE; kc += GCHUNK) {
    __syncthreads();
    // stage: arithmetic masking (no EXEC branching around the load)
    for (int i = tid; i < GCHUNK * GPAD; i += 256) {
      int r = i / GPAD, c = i % GPAD;
      int cc = (c < DIM) ? c : 0;
      float msk = (c < DIM) ? 1.f : 0.f;
      se[i] = E[(n0 + kc + r) * DIM + cc] * msk;
    }
    __syncthreads();
#pragma unroll
    for (int t = 0; t < 7; ++t) {
      int tileIdx = wave * 7 + t;                  // 0..55, each 0..48 once
      int tile = (tileIdx < 49) ? tileIdx : (tileIdx - 49);
      int e0 = (tile % 7) * 16 + lm;               // A row column in LDS
      int e1 = (tile / 7) * 16 + lm;               // B col column in LDS
#pragma unroll
      for (int ks = 0; ks < GCHUNK; ks += 4) {
        int r = ks + 2 * half;
        v2f a, b;
        a.x = se[r * GPAD + e0];
        a.y = se[(r + 1) * GPAD + e0];
        b.x = se[r * GPAD + e1];
        b.y = se[(r + 1) * GPAD + e1];
        acc[t] = __builtin_amdgcn_wmma_f32_16x16x4_f32(false, a, false, b,
                                                       (short)0, acc[t],
                                                       false, false);
      }
    }
  }
  // C/D layout: vgpr j -> M=j (lanes 0-15) / M=j+8 (lanes 16-31), N=lane&15
#pragma unroll
  for (int t = 0; t < 7; ++t) {
    int tileIdx = wave * 7 + t;
    if (tileIdx < 49) {
      int e0 = (tileIdx % 7) * 16, e1 = (tileIdx / 7) * 16;
#pragma unroll
      for (int j = 0; j < 8; ++j)
        atomicAdd(&G[(e0 + j + 8 * half) * GPAD + e1 + lm], acc[t][j]);
    }
  }
}

// ---------------------------------------------------------------------------
// Column sums S[e] = sum_n E[n,e].  grid=(100 row-chunks, 2 matrices).
__global__ void k_colsum(const float* __restrict__ Ea, const float* __restrict__ Eb,
                         float* __restrict__ Sa, float* __restrict__ Sb) {
  const float* E = blockIdx.y ? Eb : Ea;
  float*       S = blockIdx.y ? Sb : Sa;
  int t = threadIdx.x;
  if (t >= DIM) return;
  int n0 = blockIdx.x * 500;
  float s = 0.f;
  for (int n = 0; n < 500; ++n) s += E[(n0 + n) * DIM + t];
  atomicAdd(&S[t], s);
}

// ---------------------------------------------------------------------------
// Tiny per-batch chain: gather -> bn2 -> (x @ Wm_b) -> bn2 -> Mobius -> t0,t1.
__global__ void k_batch(const int* __restrict__ h_idx, const int* __restrict__ r_idx,
                        const float* __restrict__ Ea, const float* __restrict__ Eb,
                        const float* __restrict__ Rre, const float* __restrict__ Rim,
                        const float* __restrict__ WmRe, const float* __restrict__ WmIm,
                        const float* __restrict__ g0r, const float* __restrict__ b0r,
                        const float* __restrict__ g1r, const float* __restrict__ b1r,
                        const float* __restrict__ g0i, const float* __restrict__ b0i,
                        const float* __restrict__ g1i, const float* __restrict__ b1i,
                        float* __restrict__ t0, float* __restrict__ t1) {
  __shared__ float ha[1600], hb[1600], xa[1600], xb[1600], ta[1600], tb[1600];
  int tid = threadIdx.x;
  for (int i = tid; i < 1600; i += 256) {
    int b = i / DIM, e = i % DIM;
    ha[i] = Ea[h_idx[b] * DIM + e];
    hb[i] = Eb[h_idx[b] * DIM + e];
  }
  __syncthreads();
  if (tid < 200) {                                  // bn2 #1
    int e = tid % DIM; bool im = tid >= DIM;
    const float* src = im ? hb : ha;  float* dst = im ? xb : xa;
    const float* g = im ? g0i : g0r;  const float* be = im ? b0i : b0r;
    float mean = 0.f, m2 = 0.f;
    for (int b = 0; b < BATCH; ++b) { float v = src[b * DIM + e]; mean += v; m2 += v * v; }
    mean *= (1.f / BATCH); m2 *= (1.f / BATCH);
    float rstd = rsqrtf(m2 - mean * mean + EPS);
    for (int b = 0; b < BATCH; ++b)
      dst[b * DIM + e] = (src[b * DIM + e] - mean) * rstd * g[e] + be[e];
  }
  __syncthreads();
  for (int i = tid; i < 1600; i += 256) {           // per-batch matmul
    int b = i / DIM, f = i % DIM;
    float sa = 0.f, sb = 0.f;
    for (int e = 0; e < DIM; ++e) {
      sa = fmaf(xa[b * DIM + e], WmRe[b * DIM * DIM + e * DIM + f], sa);
      sb = fmaf(xb[b * DIM + e], WmIm[b * DIM * DIM + e * DIM + f], sb);
    }
    ta[i] = sa; tb[i] = sb;
  }
  __syncthreads();
  if (tid < 200) {                                  // bn2 #2
    int e = tid % DIM; bool im = tid >= DIM;
    float* arr = im ? tb : ta;
    const float* g = im ? g1i : g1r;  const float* be = im ? b1i : b1r;
    float mean = 0.f, m2 = 0.f;
    for (int b = 0; b < BATCH; ++b) { float v = arr[b * DIM + e]; mean += v; m2 += v * v; }
    mean *= (1.f / BATCH); m2 *= (1.f / BATCH);
    float rstd = rsqrtf(m2 - mean * mean + EPS);
    for (int b = 0; b < BATCH; ++b)
      arr[b * DIM + e] = (arr[b * DIM + e] - mean) * rstd * g[e] + be[e];
  }
  __syncthreads();
  for (int i = tid; i < 1600; i += 256) {           // Mobius transform
    int b = i / DIM, f = i % DIM;
    int rbase = r_idx[b] * 4 * DIM + f;
    float h0 = ta[i], h1 = tb[i];
    float ra0 = Rre[rbase],           ra1 = Rim[rbase];
    float rb0 = Rre[rbase + DIM],     rb1 = Rim[rbase + DIM];
    float rc0 = Rre[rbase + 2 * DIM], rc1 = Rim[rbase + 2 * DIM];
    float rd0 = Rre[rbase + 3 * DIM], rd1 = Rim[rbase + 3 * DIM];
    float top0 = h0 * ra0 - h1 * ra1 + rb0;
    float top1 = h0 * ra1 + h1 * ra0 + rb1;
    float bot0 = h0 * rc0 - h1 * rc1 + rd0;
    float bot1 = h0 * rc1 + h1 * rc0 + rd1;
    float den = bot0 * bot0 + bot1 * bot1;
    t0[i] = (top0 * bot0 + top1 * bot1) / den;
    t1[i] = (top1 * bot0 - top0 * bot1) / den;
  }
}

// ---------------------------------------------------------------------------
// q[b,f] = sum_{e,e'} Wm[b,e,f] G[e,e'] Wm[b,e',f]
__global__ void k_qdiag(const float* __restrict__ Ga, const float* __restrict__ Gb,
                        const float* __restrict__ WmRe, const float* __restrict__ WmIm,
                        float* __restrict__ qa, float* __restrict__ qb) {
  int id = blockIdx.x * 256 + threadIdx.x;
  if (id >= 2 * BATCH * DIM) return;
  int mat = id / (BATCH * DIM);
  int rem = id % (BATCH * DIM);
  int b = rem / DIM, f = rem % DIM;
  const float* G = mat ? Gb : Ga;
  const float* M = mat ? WmIm : WmRe;
  float acc = 0.f;
  for (int e = 0; e < DIM; ++e) {
    float inner = 0.f;
    for (int e2 = 0; e2 < DIM; ++e2)
      inner = fmaf(G[e * GPAD + e2], M[b * DIM * DIM + e2 * DIM + f], inner);
    acc = fmaf(M[b * DIM * DIM + e * DIM + f], inner, acc);
  }
  (mat ? qb : qa)[rem] = acc;
}

// ---------------------------------------------------------------------------
// bn3 stats -> fold into v[b,e] and c[b].
__global__ void k_finalize(const float* __restrict__ SWa, const float* __restrict__ SWb,
                           const float* __restrict__ Sa,  const float* __restrict__ Sb,
                           const float* __restrict__ qa,  const float* __restrict__ qb,
                           const float* __restrict__ t0,  const float* __restrict__ t1,
                           const float* __restrict__ gEa, const float* __restrict__ bEa,
                           const float* __restrict__ gEb, const float* __restrict__ bEb,
                           const float* __restrict__ WmRe, const float* __restrict__ WmIm,
                           float* __restrict__ va, float* __restrict__ vb,
                           float* __restrict__ cvec) {
  __shared__ float mean_a[DIM], rstd_a[DIM], mean_b[DIM], rstd_b[DIM];
  __shared__ float ua[BATCH * DIM], ub[BATCH * DIM];
  int tid = threadIdx.x;   // 128
  if (tid < DIM) {
    int f = tid;
    float sa = 0.f, sb = 0.f;
    for (int e = 0; e < DIM; ++e) {
      sa = fmaf(SWa[e * DIM + f], Sa[e], sa);
      sb = fmaf(SWb[e * DIM + f], Sb[e], sb);
    }
    const float inv = 1.f / BN_CNT;
    float ma = sa * inv, mb = sb * inv;
    float qsa = 0.f, qsb = 0.f;
    for (int b = 0; b < BATCH; ++b) { qsa += qa[b * DIM + f]; qsb += qb[b * DIM + f]; }
    mean_a[f] = ma; mean_b[f] = mb;
    rstd_a[f] = rsqrtf(qsa * inv - ma * ma + EPS);
    rstd_b[f] = rsqrtf(qsb * inv - mb * mb + EPS);
  }
  __syncthreads();
  for (int i = tid; i < BATCH * DIM; i += 128) {
    int f = i % DIM;
    ua[i] = t0[i] * rstd_a[f] * gEa[f];
    ub[i] = t1[i] * rstd_b[f] * gEb[f];
  }
  __syncthreads();
  if (tid < BATCH) {
    int b = tid; float cb = 0.f;
    for (int f = 0; f < DIM; ++f) {
      cb += t0[b * DIM + f] * (bEa[f] - mean_a[f] * rstd_a[f] * gEa[f]);
      cb += t1[b * DIM + f] * (bEb[f] - mean_b[f] * rstd_b[f] * gEb[f]);
    }
    cvec[b] = cb;
  }
  for (int i = tid; i < 2 * BATCH * DIM; i += 128) {
    int mat = i / (BATCH * DIM);
    int rem = i % (BATCH * DIM);
    int b = rem / DIM, e = rem % DIM;
    const float* M = mat ? WmIm : WmRe;
    const float* u = mat ? ub : ua;
    float s = 0.f;
    for (int f = 0; f < DIM; ++f)
      s = fmaf(M[b * DIM * DIM + e * DIM + f], u[b * DIM + f], s);
    (mat ? vb : va)[rem] = s;
  }
}

// ---------------------------------------------------------------------------
// score[b,n] = sigmoid(sum_e va[b,e]*Ea[n,e] + vb[b,e]*Eb[n,e] + c[b])
// blockDim=160 (5 waves) * 625 blocks = 3125 tiles exactly -> no guard, EXEC
// is all-ones for every WMMA. Operands loaded as aligned v2f (b64).
__global__ void k_score(const float* __restrict__ Ea, const float* __restrict__ Eb,
                        const float* __restrict__ va, const float* __restrict__ vb,
                        const float* __restrict__ cvec, float* __restrict__ out) {
  __shared__ float sv[2 * BATCH * DIM];
  __shared__ float sc[BATCH];
  for (int i = threadIdx.x; i < BATCH * DIM; i += 160) {
    sv[i] = va[i];
    sv[BATCH * DIM + i] = vb[i];
  }
  if (threadIdx.x < BATCH) sc[threadIdx.x] = cvec[threadIdx.x];
  __syncthreads();
  int wave = threadIdx.x >> 5;
  int lane = threadIdx.x & 31;
  int tile = blockIdx.x * 5 + wave;       // 0..3124, exact
  int n0 = tile * 16;
  int half = lane >> 4, lm = lane & 15;
  int row = n0 + lm;                      // E row feeding B column lm
  v8f acc = {};
#pragma unroll
  for (int kc = 0; kc < 25; ++kc) {       // real part: va @ Ea^T
    int k0 = kc * 4 + 2 * half;           // even -> 8B aligned
    v2f a = *(const v2f*)(&sv[lm * DIM + k0]);
    v2f b = *(const v2f*)(&Ea[row * DIM + k0]);
    acc = __builtin_amdgcn_wmma_f32_16x16x4_f32(false, a, false, b,
                                                (short)0, acc, false, false);
  }
#pragma unroll
  for (int kc = 0; kc < 25; ++kc) {       // imag part: vb @ Eb^T
    int k0 = kc * 4 + 2 * half;
    v2f a = *(const v2f*)(&sv[BATCH * DIM + lm * DIM + k0]);
    v2f b = *(const v2f*)(&Eb[row * DIM + k0]);
    acc = __builtin_amdgcn_wmma_f32_16x16x4_f32(false, a, false, b,
                                                (short)0, acc, false, false);
  }
#pragma unroll
  for (int j = 0; j < 8; ++j) {
    int m = j + 8 * half;
    float val = acc[j] + sc[m];
    out[m * NENT + n0 + lm] = 1.0f / (1.0f + __expf(-val));
  }
}

// ---------------------------------------------------------------------------
extern "C" void kernel_launch(void* const* d_in, const int* in_sizes, int n_in,
                              void* d_out, int out_size, void* d_ws, size_t ws_size,
                              hipStream_t stream) {
  (void)in_sizes; (void)n_in; (void)out_size; (void)ws_size;
  const int*   h_idx = (const int*)d_in[0];
  const int*   r_idx = (const int*)d_in[1];
  const float* Ea    = (const float*)d_in[2];
  const float* Eb    = (const float*)d_in[3];
  const float* Rre   = (const float*)d_in[4];
  const float* Rim   = (const float*)d_in[5];
  const float* R2    = (const float*)d_in[6];
  const float* Wre   = (const float*)d_in[7];
  const float* Wim   = (const float*)d_in[8];
  const float* g0r = (const float*)d_in[9],  *b0r = (const float*)d_in[10];
  const float* g1r = (const float*)d_in[11], *b1r = (const float*)d_in[12];
  const float* g0i = (const float*)d_in[13], *b0i = (const float*)d_in[14];
  const float* g1i = (const float*)d_in[15], *b1i = (const float*)d_in[16];
  const float* gEa = (const float*)d_in[17], *bEa = (const float*)d_in[18];
  const float* gEb = (const float*)d_in[19], *bEb = (const float*)d_in[20];
  float* out = (float*)d_out;

  float* ws   = (float*)d_ws;
  float* WmRe = ws + WS_WMRE;  float* WmIm = ws + WS_WMIM;
  float* SWa  = ws + WS_SWA;   float* SWb  = ws + WS_SWB;
  float* Ga   = ws + WS_GA;    float* Gb   = ws + WS_GB;
  float* Sa   = ws + WS_SA;    float* Sb   = ws + WS_SB;
  float* qa   = ws + WS_QA;    float* qb   = ws + WS_QB;
  float* t0   = ws + WS_T0;    float* t1   = ws + WS_T1;
  float* va   = ws + WS_VA;    float* vb   = ws + WS_VB;
  float* cvec = ws + WS_CV;

  k_zero<<<(WS_ZERO_CNT + 255) / 256, 256, 0, stream>>>(ws + WS_ZERO_BEG, WS_ZERO_CNT);
  k_build_wm<<<(DIM * DIM + 255) / 256, 256, 0, stream>>>(r_idx, R2, Wre, Wim,
                                                          WmRe, WmIm, SWa, SWb);
  k_gram<<<dim3(25, 1, 2), 256, 0, stream>>>(Ea, Eb, Ga, Gb);
  k_colsum<<<dim3(100, 2), 128, 0, stream>>>(Ea, Eb, Sa, Sb);
  k_batch<<<1, 256, 0, stream>>>(h_idx, r_idx, Ea, Eb, Rre, Rim, WmRe, WmIm,
                                 g0r, b0r, g1r, b1r, g0i, b0i, g1i, b1i, t0, t1);
  k_qdiag<<<(2 * BATCH * DIM + 255) / 256, 256, 0, stream>>>(Ga, Gb, WmRe, WmIm, qa, qb);
  k_finalize<<<1, 128, 0, stream>>>(SWa, SWb, Sa, Sb, qa, qb, t0, t1,
                                    gEa, bEa, gEb, bEb, WmRe, WmIm, va, vb, cvec);
  k_score<<<625, 160, 0, stream>>>(Ea, Eb, va, vb, cvec, out);
}